// UndistortPatch_67628555043543
// MI455X (gfx1250) — compile-verified
//
#include <hip/hip_runtime.h>
#include <hip/hip_bf16.h>
#include <stdint.h>

#ifndef __has_builtin
#define __has_builtin(x) 0
#endif

// Problem geometry (fixed by the reference).
#define FM_B      4
#define FM_C      128
#define FM_H      256
#define FM_W      256
#define NPATCH    256
#define PATCH_PIX 256          // 16*16 pixels per patch
#define C_PER_WG  16           // channels handled per workgroup
#define CHUNKS    (FM_C / C_PER_WG)   // 8

#if defined(__AMDGCN__)
typedef int v4i_t __attribute__((ext_vector_type(4)));
typedef v4i_t __attribute__((address_space(1))) g_v4i;   // global
typedef v4i_t __attribute__((address_space(3))) l_v4i;   // LDS
#endif

static __device__ __forceinline__ int clampi(int v, int hi) {
    return v < 0 ? 0 : (v > hi ? hi : v);
}

// One thread = one patch pixel. Weights/offsets computed once, reused across
// C_PER_WG channel planes. Output stores are non-temporal (TH=NT) so the
// 128 MB output stream does not evict the L2-resident feature map.
__global__ __launch_bounds__(PATCH_PIX)
void undistort_patch_kernel(const float* __restrict__ fm,
                            const float* __restrict__ grid,
                            float* __restrict__ out)
{
    const int p     = threadIdx.x;            // pixel within patch, 0..255
    const int n     = blockIdx.x;             // patch index, 0..255
    const int bz    = blockIdx.y;             // b * CHUNKS + chunk
    const int b     = bz >> 3;                // / CHUNKS
    const int chunk = bz & (CHUNKS - 1);
    const int c0    = chunk * C_PER_WG;

    // ---- Stage this patch's grid (256 x float2 = 2 KB) into LDS ----------
    __shared__ float2 sgrid[PATCH_PIX];
    const float* gsrc = grid + (size_t)n * PATCH_PIX * 2;

#if defined(__AMDGCN__) && \
    __has_builtin(__builtin_amdgcn_global_load_async_to_lds_b128) && \
    __has_builtin(__builtin_amdgcn_s_wait_asynccnt)
    // CDNA5 async global->LDS DMA: 128 lanes x 16B = 2 KB, tracked on ASYNCcnt.
    if (p < PATCH_PIX / 2) {
        g_v4i* src_g = (g_v4i*)(uintptr_t)(gsrc + p * 4);
        l_v4i* dst_l = (l_v4i*)((void*)&sgrid[p * 2]);
        __builtin_amdgcn_global_load_async_to_lds_b128(src_g, dst_l, 0, 0);
    }
    __builtin_amdgcn_s_wait_asynccnt(0);
    __syncthreads();
#else
    sgrid[p] = ((const float2*)gsrc)[p];
    __syncthreads();
#endif

    // ---- Per-pixel bilinear setup (done once, reused for 16 planes) -----
    const float2 g = sgrid[p];
    const float gx = (g.x + 1.0f) * 0.5f * (float)(FM_W - 1);
    const float gy = (g.y + 1.0f) * 0.5f * (float)(FM_H - 1);
    const float x0f = floorf(gx);
    const float y0f = floorf(gy);
    const float wx1 = gx - x0f, wx0 = 1.0f - wx1;
    const float wy1 = gy - y0f, wy0 = 1.0f - wy1;
    const int x0 = (int)x0f, x1 = x0 + 1;
    const int y0 = (int)y0f, y1 = y0 + 1;

    const bool vx0 = (x0 >= 0) && (x0 <= FM_W - 1);
    const bool vx1 = (x1 >= 0) && (x1 <= FM_W - 1);
    const bool vy0 = (y0 >= 0) && (y0 <= FM_H - 1);
    const bool vy1 = (y1 >= 0) && (y1 <= FM_H - 1);

    const int xc0 = clampi(x0, FM_W - 1), xc1 = clampi(x1, FM_W - 1);
    const int yc0 = clampi(y0, FM_H - 1), yc1 = clampi(y1, FM_H - 1);

    // Zero-padding semantics: clamped gather index, weight masked to 0.
    const float w00 = (vx0 && vy0) ? wx0 * wy0 : 0.0f;
    const float w10 = (vx1 && vy0) ? wx1 * wy0 : 0.0f;
    const float w01 = (vx0 && vy1) ? wx0 * wy1 : 0.0f;
    const float w11 = (vx1 && vy1) ? wx1 * wy1 : 0.0f;

    const int off00 = yc0 * FM_W + xc0;
    const int off10 = yc0 * FM_W + xc1;
    const int off01 = yc1 * FM_W + xc0;
    const int off11 = yc1 * FM_W + xc1;

    const size_t HW = (size_t)FM_H * FM_W;
    const float* __restrict__ plane0 =
        fm + ((size_t)b * FM_C + (size_t)c0) * HW;
    float* __restrict__ outp =
        out + (((size_t)b * NPATCH + (size_t)n) * FM_C + (size_t)c0) * PATCH_PIX + p;

    // ---- Channel loop: 4 gathers (RT, L2-resident) + 1 NT store ----------
#pragma unroll 4
    for (int c = 0; c < C_PER_WG; ++c) {
        const float* __restrict__ pl = plane0 + (size_t)c * HW;
        const float f00 = pl[off00];
        const float f10 = pl[off10];
        const float f01 = pl[off01];
        const float f11 = pl[off11];
        // Same summation order as the reference.
        const float v = f00 * w00 + f10 * w10 + f01 * w01 + f11 * w11;
        __builtin_nontemporal_store(v, outp + (size_t)c * PATCH_PIX);
    }
}

extern "C" void kernel_launch(void* const* d_in, const int* in_sizes, int n_in,
                              void* d_out, int out_size, void* d_ws, size_t ws_size,
                              hipStream_t stream) {
    const float* fm   = (const float*)d_in[0];  // [4,128,256,256] f32
    const float* grid = (const float*)d_in[1];  // [256,16,16,2]   f32
    float* out = (float*)d_out;                 // [4,256,128,16,16] f32

    dim3 grd(NPATCH, FM_B * CHUNKS);   // 256 x 32 = 8192 workgroups
    dim3 blk(PATCH_PIX);               // 256 threads = 8 wave32s
    undistort_patch_kernel<<<grd, blk, 0, stream>>>(fm, grid, out);
}